// GAT_30605936951896
// MI455X (gfx1250) — compile-verified
//
#include <hip/hip_runtime.h>
#include <hip/hip_bf16.h>

typedef __attribute__((ext_vector_type(2))) float v2f;
typedef __attribute__((ext_vector_type(8))) float v8f;

#define NEG_SLOPE 0.2f

// ---------------------------------------------------------------------------
// Full-precision WMMA GEMM: D[M,Nc] = A[M,K] @ B[K,Nc], all fp32 row-major.
// One 16x16 output tile per wave32; K stepped by 4 via V_WMMA_F32_16X16X4_F32.
// ---------------------------------------------------------------------------
__global__ void __launch_bounds__(256)
gat_wmma_gemm_f32(const float* __restrict__ A, const float* __restrict__ B,
                  float* __restrict__ D, int M, int K, int Nc) {
    const int wave  = (blockIdx.x * blockDim.x + threadIdx.x) >> 5;
    const int lane  = threadIdx.x & 31;
    const int tilesN = Nc >> 4;
    const int tm = wave / tilesN;
    const int tn = wave % tilesN;
    if (tm * 16 >= M) return;                       // wave-uniform guard

    const int mrow  = tm * 16 + (lane & 15);
    const int ncol  = tn * 16 + (lane & 15);
    const int khalf = (lane >> 4);                  // 0 or 1
    const int mclamp = (mrow < M) ? mrow : (M - 1);

    v8f acc = {};
    for (int k = 0; k < K; k += 4) {
        // A fragment: lane holds A[m][k + 2*khalf + {0,1}] -> contiguous pair
        const float* ap = A + (size_t)mclamp * K + k + 2 * khalf;
        v2f a; a.x = ap[0]; a.y = ap[1];
        // B fragment: lane holds B[k + 2*khalf + {0,1}][n]
        const float* bp = B + (size_t)(k + 2 * khalf) * Nc + ncol;
        v2f b; b.x = bp[0]; b.y = bp[Nc];
        acc = __builtin_amdgcn_wmma_f32_16x16x4_f32(
                  false, a, false, b, (short)0, acc, false, false);
    }

    // D fragment: acc[r] -> D[tm*16 + r + 8*khalf][ncol]
    const int rbase = tm * 16 + 8 * khalf;
#pragma unroll
    for (int r = 0; r < 8; ++r) {
        if (rbase + r < M)
            D[(size_t)(rbase + r) * Nc + ncol] = acc[r];
    }
}

// ---------------------------------------------------------------------------
// Per-(node, head) attention dots: a_src[n,h] = <h[n,h,:], att_src[h,:]>, same dst
// ---------------------------------------------------------------------------
__global__ void gat_attn_dots(const float* __restrict__ h,
                              const float* __restrict__ att_s,
                              const float* __restrict__ att_d,
                              float* __restrict__ asrc, float* __restrict__ adst,
                              int N) {
    int tid = blockIdx.x * blockDim.x + threadIdx.x;
    if (tid >= N * 4) return;
    const int n = tid >> 2, hh = tid & 3;
    const float* hp = h + (size_t)n * 256 + hh * 64;
    const float* as = att_s + hh * 64;
    const float* ad = att_d + hh * 64;
    float s = 0.f, d = 0.f;
#pragma unroll 8
    for (int c = 0; c < 64; ++c) { s += hp[c] * as[c]; d += hp[c] * ad[c]; }
    asrc[tid] = s;
    adst[tid] = d;
}

__device__ __forceinline__ void gat_get_edge(const long long* __restrict__ ei,
                                             int E, int eidx, int& s, int& d) {
    if (eidx < E) { s = (int)ei[eidx]; d = (int)ei[(size_t)E + eidx]; }
    else          { s = d = eidx - E; }              // appended self-loops
}

// ---------------------------------------------------------------------------
// Edge scores: w = exp(leaky_relu(a_src[src]+a_dst[dst])); denom[dst] += w.
// (softmax is shift-invariant; scores are O(1) so the max-shift is skipped)
// ---------------------------------------------------------------------------
__global__ void gat_edge_weights(const long long* __restrict__ ei, int E, int Etot,
                                 const float* __restrict__ asrc,
                                 const float* __restrict__ adst,
                                 float* __restrict__ wbuf,
                                 float* __restrict__ denom) {
    int tid = blockIdx.x * blockDim.x + threadIdx.x;
    if (tid >= Etot * 4) return;
    const int hh = tid & 3, eidx = tid >> 2;
    int s, d; gat_get_edge(ei, E, eidx, s, d);
    float e = asrc[s * 4 + hh] + adst[d * 4 + hh];
    e = (e > 0.f) ? e : NEG_SLOPE * e;
    float w = __expf(e);
    wbuf[tid] = w;                                    // tid == eidx*4+hh
    unsafeAtomicAdd(&denom[d * 4 + hh], w);           // global_atomic_add_f32
}

// ---------------------------------------------------------------------------
// Edge scatter: S[dst,h,c] += w[e,h] * h[src,h,c]. 256 threads per edge.
// ---------------------------------------------------------------------------
__global__ void gat_edge_scatter(const long long* __restrict__ ei, int E, int Etot,
                                 const float* __restrict__ h,
                                 const float* __restrict__ wbuf,
                                 float* __restrict__ S) {
    const int eidx = blockIdx.x;
    if (eidx >= Etot) return;
    const int idx = threadIdx.x;                      // 0..255 -> h*64+c
    int s, d; gat_get_edge(ei, E, eidx, s, d);
    const int hh = idx >> 6;
    const float w = wbuf[eidx * 4 + hh];
    const float val = w * h[(size_t)s * 256 + idx];
    unsafeAtomicAdd(&S[(size_t)d * 256 + idx], val);
}

// ---------------------------------------------------------------------------
// Finalize: out[n,c] = relu( mean_h( S[n,h,c]/denom[n,h] ) + bias[c] )
// ---------------------------------------------------------------------------
__global__ void gat_finalize(const float* __restrict__ S,
                             const float* __restrict__ denom,
                             const float* __restrict__ bias,
                             float* __restrict__ out, int N) {
    int tid = blockIdx.x * blockDim.x + threadIdx.x;
    if (tid >= N * 64) return;
    const int n = tid >> 6, c = tid & 63;
    float acc = 0.f;
#pragma unroll
    for (int hh = 0; hh < 4; ++hh)
        acc += S[(size_t)n * 256 + hh * 64 + c] / denom[n * 4 + hh];
    float v = acc * 0.25f + bias[c];
    out[tid] = (v > 0.f) ? v : 0.f;
}

// ---------------------------------------------------------------------------
// MLP head (tiny; VALU)
// ---------------------------------------------------------------------------
__global__ void gat_mlp1(const float* __restrict__ in, const float* __restrict__ W,
                         const float* __restrict__ b, float* __restrict__ out, int N) {
    int tid = blockIdx.x * blockDim.x + threadIdx.x;
    if (tid >= N * 32) return;
    const int n = tid >> 5, j = tid & 31;
    float acc = b[j];
#pragma unroll 8
    for (int k = 0; k < 64; ++k) acc += in[(size_t)n * 64 + k] * W[k * 32 + j];
    out[tid] = (acc > 0.f) ? acc : 0.f;
}

__global__ void gat_mlp2(const float* __restrict__ in, const float* __restrict__ W,
                         const float* __restrict__ b, float* __restrict__ out, int N) {
    int tid = blockIdx.x * blockDim.x + threadIdx.x;
    if (tid >= N * 2) return;
    const int n = tid >> 1, j = tid & 1;
    float acc = b[j];
#pragma unroll
    for (int k = 0; k < 32; ++k) acc += in[(size_t)n * 32 + k] * W[k * 2 + j];
    out[tid] = acc;
}

// ---------------------------------------------------------------------------
extern "C" void kernel_launch(void* const* d_in, const int* in_sizes, int n_in,
                              void* d_out, int out_size, void* d_ws, size_t ws_size,
                              hipStream_t stream) {
    const float*     x   = (const float*)d_in[0];
    const long long* ei  = (const long long*)d_in[1];
    const float*     W1  = (const float*)d_in[2];
    const float*     as1 = (const float*)d_in[3];
    const float*     ad1 = (const float*)d_in[4];
    const float*     b1  = (const float*)d_in[5];
    const float*     W2  = (const float*)d_in[6];
    const float*     as2 = (const float*)d_in[7];
    const float*     ad2 = (const float*)d_in[8];
    const float*     b2  = (const float*)d_in[9];
    const float*     Wm1 = (const float*)d_in[10];
    const float*     bm1 = (const float*)d_in[11];
    const float*     Wm2 = (const float*)d_in[12];
    const float*     bm2 = (const float*)d_in[13];

    const int N    = in_sizes[0] / 128;   // 50000
    const int E    = in_sizes[1] / 2;     // 800000
    const int Etot = E + N;               // + self loops

    // workspace carve-out (floats)
    float* ws = (float*)d_ws;
    size_t off = 0;
    float* hbuf  = ws + off; off += (size_t)N * 256;   // layer GEMM output (reused)
    float* Sbuf  = ws + off; off += (size_t)N * 256;   // scatter accumulator (reused)
    float* asrc  = ws + off; off += (size_t)N * 4;
    float* adst  = ws + off; off += (size_t)N * 4;
    float* denom = ws + off; off += (size_t)N * 4;
    float* node  = ws + off; off += (size_t)N * 64;    // per-layer node features
    float* wbuf  = ws + off; off += (size_t)Etot * 4;  // edge softmax numerators
    float* tbuf  = ws + off; off += (size_t)N * 32;    // MLP hidden

    const int TPB = 256;
    const int wavesPerBlk = TPB / 32;
    const int tilesM = (N + 15) / 16;
    const int gemmWaves = tilesM * (256 / 16);         // Nc=256 -> 16 col tiles
    const int gemmBlocks = (gemmWaves + wavesPerBlk - 1) / wavesPerBlk;

    // ---------------- layer 1 ----------------
    gat_wmma_gemm_f32<<<gemmBlocks, TPB, 0, stream>>>(x, W1, hbuf, N, 128, 256);
    gat_attn_dots<<<(N * 4 + TPB - 1) / TPB, TPB, 0, stream>>>(hbuf, as1, ad1, asrc, adst, N);
    hipMemsetAsync(Sbuf, 0, (size_t)N * 256 * sizeof(float), stream);
    hipMemsetAsync(denom, 0, (size_t)N * 4 * sizeof(float), stream);
    gat_edge_weights<<<(Etot * 4 + TPB - 1) / TPB, TPB, 0, stream>>>(ei, E, Etot, asrc, adst, wbuf, denom);
    gat_edge_scatter<<<Etot, 256, 0, stream>>>(ei, E, Etot, hbuf, wbuf, Sbuf);
    gat_finalize<<<(N * 64 + TPB - 1) / TPB, TPB, 0, stream>>>(Sbuf, denom, b1, node, N);

    // ---------------- layer 2 ----------------
    gat_wmma_gemm_f32<<<gemmBlocks, TPB, 0, stream>>>(node, W2, hbuf, N, 64, 256);
    gat_attn_dots<<<(N * 4 + TPB - 1) / TPB, TPB, 0, stream>>>(hbuf, as2, ad2, asrc, adst, N);
    hipMemsetAsync(Sbuf, 0, (size_t)N * 256 * sizeof(float), stream);
    hipMemsetAsync(denom, 0, (size_t)N * 4 * sizeof(float), stream);
    gat_edge_weights<<<(Etot * 4 + TPB - 1) / TPB, TPB, 0, stream>>>(ei, E, Etot, asrc, adst, wbuf, denom);
    gat_edge_scatter<<<Etot, 256, 0, stream>>>(ei, E, Etot, hbuf, wbuf, Sbuf);
    gat_finalize<<<(N * 64 + TPB - 1) / TPB, TPB, 0, stream>>>(Sbuf, denom, b2, node, N);

    // ---------------- MLP head ----------------
    gat_mlp1<<<(N * 32 + TPB - 1) / TPB, TPB, 0, stream>>>(node, Wm1, bm1, tbuf, N);
    gat_mlp2<<<(N * 2 + TPB - 1) / TPB, TPB, 0, stream>>>(tbuf, Wm2, bm2, (float*)d_out, N);
}